// SAGEConv_23115513987258
// MI455X (gfx1250) — compile-verified
//
#include <hip/hip_runtime.h>

typedef __attribute__((ext_vector_type(2))) float v2f;
typedef __attribute__((ext_vector_type(8))) float v8f;

#define NN 100000
#define NE 800000
#define DD 128
#define LDSW 132   // padded row stride (floats): bank stride 4 -> conflict-free ds_load_b64

__device__ __forceinline__ void atomAddF(float* p, float v) {
#if defined(__HIP_PLATFORM_AMD__)
    unsafeAtomicAdd(p, v);   // lowers to global_atomic_add_f32
#else
    atomicAdd(p, v);
#endif
}

// Stage 2: per-edge scatter-add of source features into sums[dst], plus degree.
// 32 lanes per edge; each lane handles one float4 (4 features).
__global__ void __launch_bounds__(256) sage_scatter(
    const float* __restrict__ x, const int* __restrict__ src,
    const int* __restrict__ dst, float* __restrict__ sums,
    float* __restrict__ deg) {
    int t = blockIdx.x * blockDim.x + threadIdx.x;
    int edge = t >> 5;
    int lane = t & 31;
    if (edge >= NE) return;
    int s = src[edge];
    int d = dst[edge];
    if (lane == 0) atomAddF(deg + d, 1.0f);
    const float4* xr = (const float4*)(x + (size_t)s * DD);
    float4 v = xr[lane];
    float* o = sums + (size_t)d * DD + lane * 4;
    atomAddF(o + 0, v.x);
    atomAddF(o + 1, v.y);
    atomAddF(o + 2, v.z);
    atomAddF(o + 3, v.w);
}

// Stage 3: h_neigh = sums / max(deg, 1), in place.
__global__ void __launch_bounds__(256) sage_finalize(
    float* __restrict__ sums, const float* __restrict__ deg) {
    int t = blockIdx.x * blockDim.x + threadIdx.x;
    int node = t >> 5;
    int lane = t & 31;
    if (node >= NN) return;
    float dinv = 1.0f / fmaxf(deg[node], 1.0f);
    float4* p = (float4*)(sums + (size_t)node * DD) + lane;
    float4 v = *p;
    v.x *= dinv; v.y *= dinv; v.z *= dinv; v.w *= dinv;
    *p = v;
}

// Stage 4: out = relu(x @ W_self + h @ W_neigh + b) via fp32 WMMA 16x16x4.
// One wave computes one 16x16 output tile. Block = 8 waves = one M-tile row
// (tileN = wave id covers all 8 N-tiles of D=128). Grid = 6250 M-tiles.
// A tiles (x rows + h rows for this tileM) are staged once per block into
// padded LDS, removing the 8x redundant global A-loads across waves.
__global__ void __launch_bounds__(256) sage_gemm(
    const float* __restrict__ x, const float* __restrict__ hn,
    const float* __restrict__ Ws, const float* __restrict__ Wn,
    const float* __restrict__ bias, float* __restrict__ out) {
    __shared__ float sx[16 * LDSW];
    __shared__ float sh[16 * LDSW];

    const int tileM = blockIdx.x;           // 0..6249
    const int tid = threadIdx.x;

    // Cooperative stage: 16 rows x 32 float4 per matrix = 512 chunks, 256 thr.
    // 528-byte padded rows stay 16B-aligned so float4 stores are legal.
    for (int i = tid; i < 512; i += 256) {
        int r  = i >> 5;   // row 0..15
        int c4 = i & 31;   // float4 column 0..31
        const float4 vx = ((const float4*)(x  + (size_t)(tileM * 16 + r) * DD))[c4];
        const float4 vh = ((const float4*)(hn + (size_t)(tileM * 16 + r) * DD))[c4];
        ((float4*)(sx + r * LDSW))[c4] = vx;
        ((float4*)(sh + r * LDSW))[c4] = vh;
    }
    __syncthreads();

    const int lane = tid & 31;
    const int wave = tid >> 5;              // tileN: 0..7
    const int lidx = lane & 15;
    const int hh = lane >> 4;               // wave half
    const int kb = hh * 2;                  // K sub-offset {0,2}

    const int col = wave * 16 + lidx;       // B/out column for this lane
    const float* __restrict__ ax = sx + lidx * LDSW;  // A row in LDS
    const float* __restrict__ ah = sh + lidx * LDSW;

    v8f acc = {};
#pragma unroll 4
    for (int k = 0; k < DD; k += 4) {
        v2f a, bm;
        // self term: A = x tile (16x4) from LDS, B = W_self tile (4x16)
        a.x  = ax[k + kb];
        a.y  = ax[k + kb + 1];
        bm.x = Ws[(size_t)(k + kb) * DD + col];
        bm.y = Ws[(size_t)(k + kb + 1) * DD + col];
        acc = __builtin_amdgcn_wmma_f32_16x16x4_f32(
            false, a, false, bm, (short)0, acc, false, false);
        // neighbor term: A = h_neigh tile from LDS, B = W_neigh tile
        a.x  = ah[k + kb];
        a.y  = ah[k + kb + 1];
        bm.x = Wn[(size_t)(k + kb) * DD + col];
        bm.y = Wn[(size_t)(k + kb + 1) * DD + col];
        acc = __builtin_amdgcn_wmma_f32_16x16x4_f32(
            false, a, false, bm, (short)0, acc, false, false);
    }

    const float bv = bias[col];
    float* __restrict__ obase = out + (size_t)(tileM * 16) * DD + col;
#pragma unroll
    for (int r = 0; r < 8; ++r) {
        // C/D layout: VGPR r holds row r (lanes 0-15) / row r+8 (lanes 16-31)
        float v = acc[r] + bv;
        v = fmaxf(v, 0.0f);
        obase[(size_t)(r + 8 * hh) * DD] = v;
    }
}

extern "C" void kernel_launch(void* const* d_in, const int* in_sizes, int n_in,
                              void* d_out, int out_size, void* d_ws, size_t ws_size,
                              hipStream_t stream) {
    const float* x   = (const float*)d_in[0];
    const int*   src = (const int*)d_in[1];
    const int*   dst = (const int*)d_in[2];
    const float* Ws  = (const float*)d_in[3];
    const float* Wn  = (const float*)d_in[4];
    const float* b   = (const float*)d_in[5];
    float* out = (float*)d_out;

    // Workspace layout: [deg: NN floats][pad to 512B][sums: NN*DD floats]
    size_t deg_bytes = ((size_t)NN * sizeof(float) + 511) & ~(size_t)511;
    float* deg  = (float*)d_ws;
    float* sums = (float*)((char*)d_ws + deg_bytes);
    size_t zero_bytes = deg_bytes + (size_t)NN * DD * sizeof(float);

    hipMemsetAsync(d_ws, 0, zero_bytes, stream);

    {
        int threads = NE * 32;
        sage_scatter<<<(threads + 255) / 256, 256, 0, stream>>>(x, src, dst, sums, deg);
    }
    {
        int threads = NN * 32;
        sage_finalize<<<(threads + 255) / 256, 256, 0, stream>>>(sums, deg);
    }
    sage_gemm<<<NN / 16, 256, 0, stream>>>(x, sums, Ws, Wn, b, out);
}